// MPNN_CONV_64141041599113
// MI455X (gfx1250) — compile-verified
//
#include <hip/hip_runtime.h>

// Problem constants (match reference: B=128 graphs, CH=64, D=128, H=128)
#define GB   128   // graph blocks
#define GCH  64    // nodes per block
#define GD   128   // embed dim
#define GH   128   // hidden dim
#define GN   (GB * GCH)  // 8192 nodes

typedef __attribute__((ext_vector_type(2))) float v2f;
typedef __attribute__((ext_vector_type(8))) float v8f;

// ---------------------------------------------------------------------------
// Kernel 1: block mean of embeddings.
//   m[b][d] = (1/64) * sum_{i<64} emb[b*64+i][d]
// One workgroup (128 threads) per graph block; thread t owns column t.
// Loads are fully coalesced (512B per row per block).
// ---------------------------------------------------------------------------
__global__ void __launch_bounds__(GD)
gcn_block_mean(const float* __restrict__ emb, float* __restrict__ m) {
    const int b = blockIdx.x;          // 0..127
    const int d = threadIdx.x;         // 0..127
    const float* base = emb + (size_t)b * GCH * GD + d;
    float acc = 0.0f;
#pragma unroll 8
    for (int i = 0; i < GCH; ++i) {
        acc += base[i * GD];
    }
    m[b * GD + d] = acc * (1.0f / 64.0f);
}

// ---------------------------------------------------------------------------
// Kernel 2: z = leaky_relu(m @ W + bias) via V_WMMA_F32_16X16X4_F32.
// One wave per 16x16 output tile; 8x8 tiles over the 128x128 result.
// A (16x4 f32):  lane<16 -> K=0,1 in vgpr0/1 ; lane>=16 -> K=2,3 (ISA layout)
// B (4x16 f32):  mirrored: lane-half selects K pair, vgpr selects K within pair
// C/D (16x16):   lane<16 -> M=vgpr, N=lane ; lane>=16 -> M=vgpr+8, N=lane-16
// ---------------------------------------------------------------------------
__global__ void __launch_bounds__(32)
gcn_wmma_gemm(const float* __restrict__ m, const float* __restrict__ W,
              const float* __restrict__ bias, float* __restrict__ z) {
    const int tileM = blockIdx.x >> 3;     // 0..7
    const int tileN = blockIdx.x & 7;      // 0..7
    const int lane  = threadIdx.x;         // 0..31
    const int half  = lane >> 4;           // 0 or 1 -> K-pair select
    const int lm    = lane & 15;           // row (A) / col (B,C)

    const int aRow = tileM * 16 + lm;
    const int bCol = tileN * 16 + lm;

    v8f c = {};
#pragma unroll
    for (int k0 = 0; k0 < GD; k0 += 4) {
        const int ka = k0 + half * 2;
        v2f a, bm;
        a.x  = m[aRow * GD + ka];
        a.y  = m[aRow * GD + ka + 1];
        bm.x = W[(ka)     * GH + bCol];
        bm.y = W[(ka + 1) * GH + bCol];
        // 8 args: (neg_a, A, neg_b, B, c_mod, C, reuse_a, reuse_b)
        c = __builtin_amdgcn_wmma_f32_16x16x4_f32(
            false, a, false, bm, (short)0, c, false, false);
    }

    const float bv = bias[bCol];
#pragma unroll
    for (int v = 0; v < 8; ++v) {
        const int row = tileM * 16 + v + half * 8;
        float val = c[v] + bv;
        val = (val > 0.0f) ? val : 0.01f * val;   // leaky_relu, slope 0.01
        z[row * GH + bCol] = val;
    }
}

// ---------------------------------------------------------------------------
// Kernel 3: broadcast z[b][:] to the 64 nodes of block b (float4 stores).
//   out[node][h] = z[node>>6][h]
// 1M floats -> 256K float4 stores, fully coalesced; z (64KB) hits L2/WGP$.
// ---------------------------------------------------------------------------
__global__ void __launch_bounds__(256)
gcn_broadcast(const float* __restrict__ z, float* __restrict__ out) {
    const int idx = blockIdx.x * 256 + threadIdx.x;   // float4 index
    const int totalV4 = GN * (GH / 4);                // 262144
    if (idx >= totalV4) return;
    const int colg = idx & (GH / 4 - 1);              // 0..31
    const int row  = idx >> 5;                        // node 0..8191
    const int b    = row >> 6;                        // graph block
    const float4* z4 = (const float4*)z;
    float4* o4 = (float4*)out;
    o4[idx] = z4[b * (GH / 4) + colg];
}

// ---------------------------------------------------------------------------
// Launch: inputs are (embeddings[8192*128], W[128*128], bias[128], b=128, ch=64)
// ---------------------------------------------------------------------------
extern "C" void kernel_launch(void* const* d_in, const int* in_sizes, int n_in,
                              void* d_out, int out_size, void* d_ws, size_t ws_size,
                              hipStream_t stream) {
    const float* emb  = (const float*)d_in[0];
    const float* W    = (const float*)d_in[1];
    const float* bias = (const float*)d_in[2];
    float* out = (float*)d_out;

    float* m = (float*)d_ws;                              // 128*128 f32 = 64KB
    float* z = (float*)((char*)d_ws + GB * GD * sizeof(float)); // next 64KB

    gcn_block_mean<<<GB, GD, 0, stream>>>(emb, m);
    gcn_wmma_gemm<<<64, 32, 0, stream>>>(m, W, bias, z);
    const int totalV4 = GN * (GH / 4);
    gcn_broadcast<<<(totalV4 + 255) / 256, 256, 0, stream>>>(z, out);
}